// TAACTDLoss_25709674234608
// MI455X (gfx1250) — compile-verified
//
#include <hip/hip_runtime.h>
#include <hip/hip_bf16.h>

// TD-return backward scan + squared-error loss.
// T=256 timesteps, B=65536 independent columns. Pure streaming workload:
// 512 MB of traffic, ~22 us roofline at 23.3 TB/s HBM. One thread per
// column; wave32 lanes map to adjacent columns -> every load is exactly one
// 128B cacheline per wave. All traffic is single-use -> NT cache hints.
// Full-coverage global_prefetch_b8 stream (16 rows ahead, all 7 arrays)
// turns demand HBM misses into L2 hits. Prefetch addresses are a constant
// byte delta (-4 MB) from the demand-load index so they encode as immediate
// offsets off the same base register (no extra address VALU).

#ifndef __HIP_DEVICE_COMPILE__
#include <cstddef>
#endif

namespace {
constexpr int   kT     = 256;
constexpr int   kB     = 65536;
constexpr float kGamma = 0.98f;
constexpr int   kStepTypeLast = 2;
constexpr int   kBlock = 256;   // 8 wave32s per workgroup
constexpr int   kPfDist = 16;   // prefetch distance in timesteps (rows)
}  // namespace

__global__ __launch_bounds__(kBlock) void td_scan_loss_kernel(
    const float* __restrict__ reward,
    const float* __restrict__ discount,
    const float* __restrict__ value,
    const float* __restrict__ target_value,
    const int*   __restrict__ b_arr,
    const int*   __restrict__ rollout_b,
    const int*   __restrict__ step_type,
    float* __restrict__ out)
{
    const int c = blockIdx.x * kBlock + threadIdx.x;  // column index, < kB

    // acc starts at target_value[T-1, c]
    float acc = __builtin_nontemporal_load(&target_value[(kT - 1) * kB + c]);

    // Reference concatenates a zero row at t = T-1.
    __builtin_nontemporal_store(0.0f, &out[(kT - 1) * kB + c]);

    // Backward scan: t = T-2 .. 0.
    // Loads are independent of `acc`; unrolling lets the compiler hoist many
    // iterations of loads ahead of the serial fma/cndmask chain.
    #pragma unroll 8
    for (int t = kT - 2; t >= 0; --t) {
        const int r0 = t * kB + c;       // row t
        const int r1 = r0 + kB;          // row t+1
        const int rp = r0 - kPfDist * kB; // row t-16: constant -4 MB delta.
        // For t < 16 this is below the buffer start: speculative prefetch
        // (TH=0) is silently dropped on translation failure -- safe by ISA.

        const float dis = __builtin_nontemporal_load(&discount[r1]);
        const float rew = __builtin_nontemporal_load(&reward[r1]);
        const float val = __builtin_nontemporal_load(&value[r0]);
        const float tgt = __builtin_nontemporal_load(&target_value[r0]);
        const int   bi  = __builtin_nontemporal_load(&b_arr[r0]);
        const int   ri  = __builtin_nontemporal_load(&rollout_b[r0]);
        const int   si  = __builtin_nontemporal_load(&step_type[r0]);

        // Full-coverage L2 prefetch stream down the iteration direction
        // (t decreasing). Lowers to global_prefetch_b8; no LOADcnt impact,
        // encoded as immediate offsets off the same index as the loads.
        __builtin_prefetch(&reward[rp], 0, 0);
        __builtin_prefetch(&discount[rp], 0, 0);
        __builtin_prefetch(&value[rp], 0, 0);
        __builtin_prefetch(&target_value[rp], 0, 0);
        __builtin_prefetch(&b_arr[rp], 0, 0);
        __builtin_prefetch(&rollout_b[rp], 0, 0);
        __builtin_prefetch(&step_type[rp], 0, 0);

        // ret_t = acc * (discount[t+1]*gamma) + reward[t+1]
        const float ret = fmaf(acc, dis * kGamma, rew);

        // loss[t] = (ret_t - value[t])^2
        const float d = ret - val;
        __builtin_nontemporal_store(d * d, &out[r0]);

        // is_last[t] = (step_type[t]==LAST) | ((b|rollout_b) & t>0)
        const bool il = (si == kStepTypeLast) | (((bi | ri) != 0) & (t != 0));

        // acc_t = il ? target_value[t] : ret_t
        acc = il ? tgt : ret;
    }
}

extern "C" void kernel_launch(void* const* d_in, const int* in_sizes, int n_in,
                              void* d_out, int out_size, void* d_ws, size_t ws_size,
                              hipStream_t stream) {
    (void)in_sizes; (void)n_in; (void)d_ws; (void)ws_size; (void)out_size;

    // setup_inputs() order:
    const float* reward       = (const float*)d_in[0];
    const float* discount     = (const float*)d_in[1];
    const float* value        = (const float*)d_in[2];
    const float* target_value = (const float*)d_in[3];
    const int*   b_arr        = (const int*)d_in[4];
    const int*   rollout_b    = (const int*)d_in[5];
    const int*   step_type    = (const int*)d_in[6];
    float*       out          = (float*)d_out;

    const int grid = kB / kBlock;  // 256 blocks * 256 threads = 65536 = B
    td_scan_loss_kernel<<<grid, kBlock, 0, stream>>>(
        reward, discount, value, target_value, b_arr, rollout_b, step_type, out);
}